// DetailAggregateLoss_2542620639462
// MI455X (gfx1250) — compile-verified
//
#include <hip/hip_runtime.h>
#include <cstdint>

// ---------------------------------------------------------------------------
// DetailAggregateLoss for MI455X (gfx1250, wave32).
// Memory-bound: 128 MB traffic / 23.3 TB/s ~= 5.5us floor. One pass:
//  pass1: per 64x64 tile -> TDM-load gtmask tile (+1 halo) into LDS,
//         compute 3-scale thresholded Laplacians + fused target,
//         BCE terms + dice partial sums, wave32 reduce (WMMA f32 16x16x4
//         ones-matrix trick for the BCE sum), write 4 floats per block.
//  pass2: single block folds 4096 partials into the two output scalars.
// ---------------------------------------------------------------------------

#define TILE  64
#define PITCH 66          // 64 + 2 halo
#define THRV  0.1f

typedef float        v2f  __attribute__((ext_vector_type(2)));
typedef float        v8f  __attribute__((ext_vector_type(8)));
typedef unsigned int u32x4 __attribute__((ext_vector_type(4)));
typedef int          i32x4 __attribute__((ext_vector_type(4)));
typedef int          i32x8 __attribute__((ext_vector_type(8)));

__global__ __launch_bounds__(256) void detail_loss_pass1(
    const float* __restrict__ logits, const float* __restrict__ gt,
    const float* __restrict__ fuse, float* __restrict__ wsOut,
    int H, int W, int atomicMode)
{
    __shared__ float sh[PITCH * PITCH];
    __shared__ float wred[4][8];

    const int tid = threadIdx.x;
    const int bx = blockIdx.x, by = blockIdx.y, b = blockIdx.z;
    const int x0 = bx * TILE, y0 = by * TILE;

    // Pre-zero the halo ring (covers image-edge padding the DMA won't write).
    for (int i = tid; i < PITCH; i += 256) {
        sh[i] = 0.f;
        sh[(PITCH - 1) * PITCH + i] = 0.f;
        sh[i * PITCH] = 0.f;
        sh[i * PITCH + PITCH - 1] = 0.f;
    }
    __syncthreads();

    // Clamped fetch window [sy, sy+rows) x [sx, sx+fw) of the gtmask image.
    const int gx0 = x0 - 1, gy0 = y0 - 1;
    const int sx = gx0 < 0 ? 0 : gx0;
    const int sy = gy0 < 0 ? 0 : gy0;
    const int ox = sx - gx0, oy = sy - gy0;
    const int rows = min(PITCH - oy, H - sy);
    const int fw = PITCH - ox;

#if __has_builtin(__builtin_amdgcn_tensor_load_to_lds) && __has_builtin(__builtin_amdgcn_s_wait_tensorcnt)
    if (tid < 32) {  // wave 0 issues one TDM descriptor for the whole tile
        const unsigned long long ga =
            (unsigned long long)(uintptr_t)(gt + ((size_t)b * H + sy) * W + sx);
        const unsigned lds0 = (unsigned)(uintptr_t)(&sh[oy * PITCH + ox]);
        const int td0 = W - sx;   // right-edge OOB -> zeros (matches conv pad)
        const int td1 = H - sy;

        u32x4 g0;
        g0[0] = 1u;                                               // count=1, user D#
        g0[1] = lds0;                                             // lds_addr (bytes)
        g0[2] = (unsigned)ga;                                     // global_addr lo
        g0[3] = ((unsigned)(ga >> 32) & 0x01FFFFFFu) | 0x80000000u; // hi | type=2

        i32x8 g1;
        g1[0] = (2 << 16) | (1 << 19);                 // data_size=4B, iterate_enable
        g1[1] = (td0 & 0xFFFF) << 16;                  // tensor_dim0[15:0]
        g1[2] = ((td0 >> 16) & 0xFFFF) | ((td1 & 0xFFFF) << 16);
        g1[3] = ((td1 >> 16) & 0xFFFF) | (fw << 16);   // tile_dim0 = fetch width
        g1[4] = 1;                                     // tile_dim1 = 1 row / iter
        g1[5] = W;                                     // tensor_dim0_stride
        g1[6] = 0;
        g1[7] = 0;

        i32x4 g2;
        g2[0] = 0;                                     // tensor_dim2
        g2[1] = PITCH;                                 // lds_addr_increment (elems)
        g2[2] = W;                                     // global_addr_increment (elems)
        g2[3] = (rows - 1) << 16;                      // iterate_count (0 => 1x)

        i32x4 g3 = {0, 0, 0, 0};
        i32x8 g4 = {0, 0, 0, 0, 0, 0, 0, 0};           // extra group (clang-23 form)

        __builtin_amdgcn_tensor_load_to_lds(g0, g1, g2, g3, g4, 0);
        __builtin_amdgcn_s_wait_tensorcnt(0);
    }
#else
    for (int i = tid; i < rows * fw; i += 256) {
        const int r = i / fw, c = i % fw;
        const int gx = sx + c;
        float v = 0.f;
        if (gx < W) v = gt[((size_t)b * H + sy + r) * W + gx];
        sh[(oy + r) * PITCH + ox + c] = v;
    }
#endif
    __syncthreads();

    // ---- compute: each thread owns a 4x4 pixel block (4-aligned, so the
    //      stride-2/4 Laplacian centers all live inside this tile) ----
    const int tix = tid & 15, tiy = tid >> 4;
    const int px = tix * 4, py = tiy * 4;  // tile-local origin

    auto L = [&](int y, int x) -> float { return sh[(y + 1) * PITCH + (x + 1)]; };
    auto lap = [&](int y, int x) -> float {
        return 8.f * L(y, x) -
               (L(y - 1, x - 1) + L(y - 1, x) + L(y - 1, x + 1) +
                L(y, x - 1)                  + L(y, x + 1) +
                L(y + 1, x - 1) + L(y + 1, x) + L(y + 1, x + 1));
    };

    const float f0 = fuse[0], f1 = fuse[1], f2 = fuse[2];

    const float t4 = (lap(py, px) > THRV) ? 1.f : 0.f;   // stride-4 map
    float t2v[2][2];                                     // stride-2 map
    #pragma unroll
    for (int a = 0; a < 2; ++a)
        #pragma unroll
        for (int c = 0; c < 2; ++c)
            t2v[a][c] = (lap(py + 2 * a, px + 2 * c) > THRV) ? 1.f : 0.f;

    float bce = 0.f, ptv = 0.f, pv = 0.f, tv = 0.f;
    #pragma unroll
    for (int r = 0; r < 4; ++r) {
        const int y = py + r;
        const float4 xl = *reinterpret_cast<const float4*>(
            logits + ((size_t)b * H + (y0 + y)) * W + (x0 + px));   // b128 load
        const float xs[4] = {xl.x, xl.y, xl.z, xl.w};
        #pragma unroll
        for (int c = 0; c < 4; ++c) {
            const int x = px + c;
            const float t1 = (lap(y, x) > THRV) ? 1.f : 0.f;
            const float fused = f0 * t1 + f1 * t2v[r >> 1][c >> 1] + f2 * t4;
            const float tgt = (fused > THRV) ? 1.f : 0.f;
            const float xv = xs[c];
            // stable BCE-with-logits term
            bce += fmaxf(xv, 0.f) - xv * tgt + __logf(1.f + __expf(-fabsf(xv)));
            const float p = 1.f / (1.f + __expf(-xv));
            ptv += p * tgt;
            pv += p;
            tv += tgt;
        }
    }

    // ---- wave32 reductions ----
    #pragma unroll
    for (int m = 16; m >= 1; m >>= 1) {
        ptv += __shfl_xor(ptv, m, 32);
        pv  += __shfl_xor(pv,  m, 32);
        tv  += __shfl_xor(tv,  m, 32);
    }

#if __has_builtin(__builtin_amdgcn_wmma_f32_16x16x4_f32)
    // Exact full-wave sum via WMMA: A[m][k] holds lane partials (K slot 0),
    // B = ones  =>  D[m][n] = s[m] + s[m+16]. Sum the 8 D rows per lane and
    // fold the two half-waves. EXEC is all-ones here (no divergence).
    {
        v2f a;  a[0] = bce; a[1] = 0.f;
        v2f o;  o[0] = 1.f; o[1] = 1.f;
        v8f cacc = {};
        cacc = __builtin_amdgcn_wmma_f32_16x16x4_f32(
            false, a, false, o, (short)0, cacc, false, false);
        float rsum = cacc[0] + cacc[1] + cacc[2] + cacc[3] +
                     cacc[4] + cacc[5] + cacc[6] + cacc[7];
        bce = rsum + __shfl_xor(rsum, 16, 32);
    }
#else
    #pragma unroll
    for (int m = 16; m >= 1; m >>= 1) bce += __shfl_xor(bce, m, 32);
#endif

    const int wid = tid >> 5, lane = tid & 31;
    if (lane == 0) {
        wred[0][wid] = bce; wred[1][wid] = ptv;
        wred[2][wid] = pv;  wred[3][wid] = tv;
    }
    __syncthreads();

    if (tid == 0) {
        float s0 = 0.f, s1 = 0.f, s2 = 0.f, s3 = 0.f;
        #pragma unroll
        for (int w = 0; w < 8; ++w) {
            s0 += wred[0][w]; s1 += wred[1][w];
            s2 += wred[2][w]; s3 += wred[3][w];
        }
        if (atomicMode) {
            atomicAdd(&wsOut[0], s0);
            atomicAdd(&wsOut[1 + b], s1);
            atomicAdd(&wsOut[17 + b], s2);
            atomicAdd(&wsOut[33 + b], s3);
        } else {
            const size_t pidx = (((size_t)b * gridDim.y + by) * gridDim.x + bx) * 4;
            wsOut[pidx + 0] = s0; wsOut[pidx + 1] = s1;
            wsOut[pidx + 2] = s2; wsOut[pidx + 3] = s3;
        }
    }
}

__global__ __launch_bounds__(256) void detail_loss_pass2(
    const float* __restrict__ part, float* __restrict__ out,
    int nblocks, int perBatch, int B, float invN)
{
    __shared__ float red[256];
    __shared__ float dterm[64];
    const int tid = threadIdx.x;

    // global BCE sum
    float s = 0.f;
    for (int i = tid; i < nblocks; i += 256) s += part[(size_t)i * 4 + 0];
    red[tid] = s;
    __syncthreads();
    for (int st = 128; st >= 1; st >>= 1) {
        if (tid < st) red[tid] += red[tid + st];
        __syncthreads();
    }
    const float bce_loss = red[0] * invN;

    // per-batch dice sums: 16 lanes per batch
    const int bb = tid >> 4, j = tid & 15;
    float spt = 0.f, sp = 0.f, stt = 0.f;
    if (bb < B) {
        for (int k = j; k < perBatch; k += 16) {
            const size_t idx = ((size_t)bb * perBatch + k) * 4;
            spt += part[idx + 1];
            sp  += part[idx + 2];
            stt += part[idx + 3];
        }
    }
    #pragma unroll
    for (int m = 8; m >= 1; m >>= 1) {
        spt += __shfl_xor(spt, m, 32);
        sp  += __shfl_xor(sp,  m, 32);
        stt += __shfl_xor(stt, m, 32);
    }
    if (j == 0 && bb < B)
        dterm[bb] = 1.f - (2.f * spt + 1.f) / (sp + stt + 1.f);
    __syncthreads();

    if (tid == 0) {
        float d = 0.f;
        for (int i = 0; i < B; ++i) d += dterm[i];
        out[0] = bce_loss;
        out[1] = d / (float)B;
    }
}

__global__ void detail_loss_zero(float* __restrict__ acc, int n)
{
    if ((int)threadIdx.x < n) acc[threadIdx.x] = 0.f;
}

__global__ void detail_loss_final_atomic(const float* __restrict__ acc,
                                         float* __restrict__ out,
                                         int B, float invN)
{
    if (threadIdx.x == 0) {
        out[0] = acc[0] * invN;
        float d = 0.f;
        for (int i = 0; i < B; ++i)
            d += 1.f - (2.f * acc[1 + i] + 1.f) / (acc[17 + i] + acc[33 + i] + 1.f);
        out[1] = d / (float)B;
    }
}

extern "C" void kernel_launch(void* const* d_in, const int* in_sizes, int n_in,
                              void* d_out, int out_size, void* d_ws, size_t ws_size,
                              hipStream_t stream)
{
    const float* logits = (const float*)d_in[0];   // [B,1,H,W] fp32
    const float* gt     = (const float*)d_in[1];   // [B,H,W]   fp32
    const float* fuse   = (const float*)d_in[2];   // [3]       fp32
    float* out = (float*)d_out;                    // {bce, dice}

    const int H = 1024, W = 1024;
    const int B = in_sizes[1] / (H * W);

    dim3 grid(W / TILE, H / TILE, B);
    const int perBatch = grid.x * grid.y;
    const int nblocks  = perBatch * B;
    const float invN = (float)(1.0 / ((double)B * H * W));

    const size_t needDet = (size_t)nblocks * 4 * sizeof(float);
    if (ws_size >= needDet) {
        detail_loss_pass1<<<grid, 256, 0, stream>>>(logits, gt, fuse,
                                                    (float*)d_ws, H, W, 0);
        detail_loss_pass2<<<1, 256, 0, stream>>>((const float*)d_ws, out,
                                                 nblocks, perBatch, B, invN);
    } else {
        detail_loss_zero<<<1, 64, 0, stream>>>((float*)d_ws, 49);
        detail_loss_pass1<<<grid, 256, 0, stream>>>(logits, gt, fuse,
                                                    (float*)d_ws, H, W, 1);
        detail_loss_final_atomic<<<1, 32, 0, stream>>>((const float*)d_ws, out,
                                                       B, invN);
    }
}